// MyModel_61933428412445
// MI455X (gfx1250) — compile-verified
//
#include <hip/hip_runtime.h>
#include <hip/hip_bf16.h>

typedef float v2f __attribute__((ext_vector_type(2)));
typedef float v8f __attribute__((ext_vector_type(8)));

static constexpr int T_ = 2048;
static constexpr int B_ = 4096;
static constexpr int I_ = 4;
static constexpr int H_ = 4;

__device__ __forceinline__ float sigm(float x) {
  // 1 / (1 + 2^(-x * log2(e)))  -> v_exp_f32 + v_rcp_f32
  return __builtin_amdgcn_rcpf(1.0f + __builtin_amdgcn_exp2f(-1.4426950408889634f * x));
}

__device__ __forceinline__ float tanh_f(float x) {
#if __has_builtin(__builtin_amdgcn_tanh_f32)
  return __builtin_amdgcn_tanh_f32(x);
#elif __has_builtin(__builtin_amdgcn_tanhf)
  return __builtin_amdgcn_tanhf(x);
#else
  return 2.0f * sigm(2.0f * x) - 1.0f;
#endif
}

__device__ __forceinline__ v8f wmma4(v2f a, v2f b, v8f c) {
  // D(16x16) = A(16x4) * B(4x16) + C ; f32 everywhere
  return __builtin_amdgcn_wmma_f32_16x16x4_f32(false, a, false, b, (short)0, c,
                                               false, false);
}

// One wave handles 16 batch elements of one cell type.
// Orientation: D(gate M x batch N) = W(gate x K=4) * [x|h]^T(4 x 16) + bias.
// A layout (16x4 f32): lane m -> K=0,1 ; lane m+16 -> K=2,3  (rows = gates, zero-padded)
// B layout (4x16 f32): lane n -> K=0 (v0), K=1 (v1) ; lane n+16 -> K=2 (v0), K=3 (v1)
// C/D layout: VGPR r, lane l: gate M = r + 8*(l>=16), batch N = l%16
__global__ __launch_bounds__(256) void fused_rnn_lstm_gru(
    const float* __restrict__ x,
    const float* __restrict__ rWih, const float* __restrict__ rWhh,
    const float* __restrict__ rbih, const float* __restrict__ rbhh,
    const float* __restrict__ lWih, const float* __restrict__ lWhh,
    const float* __restrict__ lbih, const float* __restrict__ lbhh,
    const float* __restrict__ gWih, const float* __restrict__ gWhh,
    const float* __restrict__ gbih, const float* __restrict__ gbhh,
    float* __restrict__ out)
{
  const int  lane  = (int)(threadIdx.x & 31u);
  const int  wid   = (int)(threadIdx.x >> 5);
  const int  cell  = (int)blockIdx.y;                  // 0=rnn, 1=lstm, 2=gru
  const int  bbase = ((int)blockIdx.x * 8 + wid) * 16; // 16 batches per wave
  const bool lo    = lane < 16;
  const int  m     = lane & 15;
  const int  kb    = lo ? 0 : 2;

  const float *Wih, *Whh, *bih, *bhh;
  int G;
  if (cell == 0)      { Wih = rWih; Whh = rWhh; bih = rbih; bhh = rbhh; G = H_;     }
  else if (cell == 1) { Wih = lWih; Whh = lWhh; bih = lbih; bhh = lbhh; G = 4 * H_; }
  else                { Wih = gWih; Whh = gWhh; bih = gbih; bhh = gbhh; G = 3 * H_; }

  // A operands: torch (G, 4) row-major is exactly the 16x4 A tile (zero-pad rows >= G).
  v2f a1; a1.x = 0.f; a1.y = 0.f;
  v2f a2; a2.x = 0.f; a2.y = 0.f;
  if (m < G) {
    a1.x = Wih[m * I_ + kb];  a1.y = Wih[m * I_ + kb + 1];
    a2.x = Whh[m * H_ + kb];  a2.y = Whh[m * H_ + kb + 1];
  }

  // Bias C operands: cb1 = bih broadcast over batch columns, cb2 = bhh.
  v8f cb1, cb2;
#pragma unroll
  for (int r = 0; r < 8; ++r) {
    int M = r + (lo ? 0 : 8);
    cb1[r] = (M < G) ? bih[M] : 0.f;
    cb2[r] = (M < G) ? bhh[M] : 0.f;
  }

  // Per-lane element offset into x / out (both have inner size 4, so it is shared).
  const int    b   = bbase + m;
  const size_t off = (size_t)b * 4 + (size_t)kb;
  float* outc = out + (size_t)cell * (size_t)T_ * (size_t)B_ * (size_t)H_;

  float h[4]  = {0.f, 0.f, 0.f, 0.f};
  float cs[4] = {0.f, 0.f, 0.f, 0.f};

  if (cell == 0) {
    // ---------------- vanilla RNN: h = tanh(pre + h Whh^T + bhh) ----------------
    for (int t = 0; t < T_; ++t) {
      v2f bx = *(const v2f*)(x + (size_t)t * (B_ * I_) + off);
      v2f bh; bh.x = lo ? h[0] : h[2]; bh.y = lo ? h[1] : h[3];
      v8f d1 = wmma4(a1, bx, cb1);
      v8f d2 = wmma4(a2, bh, cb2);
#pragma unroll
      for (int j = 0; j < 4; ++j) {        // low lanes hold all 4 gates (G=4)
        float v  = tanh_f(d1[j] + d2[j]);
        float vx = __shfl_xor(v, 16, 32);  // replicate into high half for B/store
        h[j] = lo ? v : vx;
      }
      v2f hv; hv.x = lo ? h[0] : h[2]; hv.y = lo ? h[1] : h[3];
      *(v2f*)(outc + (size_t)t * (B_ * H_) + off) = hv;
    }
  } else if (cell == 1) {
    // ---------------- LSTM: gates i,f (low lane) | g,o (high lane) ----------------
    for (int t = 0; t < T_; ++t) {
      v2f bx = *(const v2f*)(x + (size_t)t * (B_ * I_) + off);
      v2f bh; bh.x = lo ? h[0] : h[2]; bh.y = lo ? h[1] : h[3];
      v8f d1 = wmma4(a1, bx, cb1);
      v8f d2 = wmma4(a2, bh, cb2);
#pragma unroll
      for (int j = 0; j < 4; ++j) {
        float sA = d1[j]     + d2[j];      // low: i_j   high: g_j
        float sB = d1[j + 4] + d2[j + 4];  // low: f_j   high: o_j
        float A  = lo ? sigm(sA) : tanh_f(sA);
        float Bv = sigm(sB);
        float Ax = __shfl_xor(A, 16, 32);
        float Bx = __shfl_xor(Bv, 16, 32);
        float si = lo ? A  : Ax;
        float tg = lo ? Ax : A;
        float sf = lo ? Bv : Bx;
        float so = lo ? Bx : Bv;
        cs[j] = sf * cs[j] + si * tg;      // state duplicated in both half-waves
        h[j]  = so * tanh_f(cs[j]);
      }
      v2f hv; hv.x = lo ? h[0] : h[2]; hv.y = lo ? h[1] : h[3];
      *(v2f*)(outc + (size_t)t * (B_ * H_) + off) = hv;
    }
  } else {
    // ---------------- GRU: gates r,z (low lane) | n (high lane) ----------------
    for (int t = 0; t < T_; ++t) {
      v2f bx = *(const v2f*)(x + (size_t)t * (B_ * I_) + off);
      v2f bh; bh.x = lo ? h[0] : h[2]; bh.y = lo ? h[1] : h[3];
      v8f d1 = wmma4(a1, bx, cb1);   // pre (+bih), gates r,z | n
      v8f d2 = wmma4(a2, bh, cb2);   // h Whh^T (+bhh) kept separate: n = tanh(pn + r*hn)
#pragma unroll
      for (int j = 0; j < 4; ++j) {
        float r_ = sigm(d1[j]     + d2[j]);       // valid in low lanes
        float z_ = sigm(d1[j + 4] + d2[j + 4]);   // valid in low lanes
        float rs = lo ? r_ : 0.f;
        float rx = __shfl_xor(rs, 16, 32);
        float rr = lo ? r_ : rx;                  // high lanes now have r
        float n_ = tanh_f(d1[j] + rr * d2[j]);    // valid in high lanes (pn + r*hn)
        float nx = __shfl_xor(n_, 16, 32);
        float N  = lo ? nx : n_;
        float zx = __shfl_xor(z_, 16, 32);
        float Z  = lo ? z_ : zx;
        h[j] = (1.f - Z) * N + Z * h[j];
      }
      v2f hv; hv.x = lo ? h[0] : h[2]; hv.y = lo ? h[1] : h[3];
      *(v2f*)(outc + (size_t)t * (B_ * H_) + off) = hv;
    }
  }
}

extern "C" void kernel_launch(void* const* d_in, const int* in_sizes, int n_in,
                              void* d_out, int out_size, void* d_ws, size_t ws_size,
                              hipStream_t stream) {
  const float* x    = (const float*)d_in[0];
  const float* rWih = (const float*)d_in[1];
  const float* rWhh = (const float*)d_in[2];
  const float* rbih = (const float*)d_in[3];
  const float* rbhh = (const float*)d_in[4];
  const float* lWih = (const float*)d_in[5];
  const float* lWhh = (const float*)d_in[6];
  const float* lbih = (const float*)d_in[7];
  const float* lbhh = (const float*)d_in[8];
  const float* gWih = (const float*)d_in[9];
  const float* gWhh = (const float*)d_in[10];
  const float* gbih = (const float*)d_in[11];
  const float* gbhh = (const float*)d_in[12];
  float* out = (float*)d_out;

  // 32 blocks x 8 waves x 16 batches = 4096 batches; grid.y = cell type.
  dim3 grid(B_ / (16 * 8), 3, 1);
  fused_rnn_lstm_gru<<<grid, dim3(256, 1, 1), 0, stream>>>(
      x, rWih, rWhh, rbih, rbhh, lWih, lWhh, lbih, lbhh, gWih, gWhh, gbih, gbhh,
      out);

  (void)in_sizes; (void)n_in; (void)out_size; (void)d_ws; (void)ws_size;
}